// MemNet_90220083020109
// MI455X (gfx1250) — compile-verified
//
#include <hip/hip_runtime.h>
#include <hip/hip_bf16.h>

#define VOCAB 50000
#define D     300
#define L     256
#define LT    8
#define B     256
#define NHOPS 6
#define NPAD  304   // D padded to a multiple of 16 for WMMA tiles

typedef float v2f __attribute__((ext_vector_type(2)));
typedef float v8f __attribute__((ext_vector_type(8)));

// ---------------------------------------------------------------------------
// Kernel 0: one-time prep.  wT[n][k] = w_lin[k][n] (zero-padded to n<304),
// b_pad[n] = b_lin[n] (zero-padded).  Transposed+padded layout makes the
// WMMA B-fragment loads contiguous b64 loads with no predication.
// ---------------------------------------------------------------------------
__global__ void k_prep(const float* __restrict__ w_lin,
                       const float* __restrict__ b_lin,
                       float* __restrict__ wT,
                       float* __restrict__ b_pad) {
    const int n = blockIdx.x;            // 0..303
    const bool nvalid = (n < D);
    for (int k = threadIdx.x; k < D; k += blockDim.x)
        wT[n * D + k] = nvalid ? w_lin[k * D + n] : 0.0f;
    if (threadIdx.x == 0)
        b_pad[n] = nvalid ? b_lin[n] : 0.0f;
}

// ---------------------------------------------------------------------------
// Kernel 1: gather + location-scale + hop-invariant attention score.
// One wave per (b,l) row.  ext_mem[b,l,:] = emb[ctx[b,l],:] * v_loc(b,l)
// mem_score[b,l] = dot(ext_mem[b,l,:], w_att[:D])   (reused by all 6 hops)
// ---------------------------------------------------------------------------
__global__ void k_gather(const float* __restrict__ emb,
                         const float* __restrict__ w_att,
                         const int*   __restrict__ ctx_ids,
                         const int*   __restrict__ ctx_len,
                         const int*   __restrict__ tgt_loc,
                         float* __restrict__ ext_mem,
                         float* __restrict__ mem_score) {
    const int lane = threadIdx.x & 31;
    const int wave = threadIdx.x >> 5;
    const int b    = blockIdx.y;
    const int l    = blockIdx.x * 8 + wave;

    const int len = ctx_len[b];
    const int loc = tgt_loc[b];
    float vloc = 0.0f;
    if (l < len)
        vloc = 1.0f - fabsf((float)(l - loc)) / (float)len;

    const int id = ctx_ids[b * L + l];
    const float* __restrict__ e = emb + (size_t)id * D;
    float* __restrict__ o = ext_mem + ((size_t)b * L + l) * D;

    float dot = 0.0f;
    for (int d = lane; d < D; d += 32) {
        float v = e[d] * vloc;
        o[d] = v;
        dot += v * w_att[d];
    }
    for (int off = 16; off; off >>= 1) dot += __shfl_xor(dot, off, 32);
    if (lane == 0) mem_score[b * L + l] = dot;
}

// ---------------------------------------------------------------------------
// Kernel 2: aspect vector = masked mean of target embeddings.
// ---------------------------------------------------------------------------
__global__ void k_aspect(const float* __restrict__ emb,
                         const int*   __restrict__ tgt_ids,
                         const int*   __restrict__ tgt_len,
                         float* __restrict__ vec) {
    const int b = blockIdx.x;
    const int d = threadIdx.x;
    if (d >= D) return;
    const int tl = tgt_len[b];
    float s = 0.0f;
    for (int t = 0; t < tl; ++t) {
        const int id = tgt_ids[b * LT + t];
        s += emb[(size_t)id * D + d];
    }
    vec[b * D + d] = s / (float)tl;
}

// ---------------------------------------------------------------------------
// Kernel 3: lin = vec @ w_lin + b_lin via V_WMMA_F32_16X16X4_F32.
// One wave per 16x16 tile; M=256 (16 tiles), N=304 (19 tiles), K=300 (75x4).
// Branch-free inner loop: one b64 load per fragment per k-step + WMMA.
// A layout (ISA 7.12.2, 32-bit A 16x4): lane%16 = M; VGPR0/1 = K + 2*(lane/16).
// B layout mirrored: lane%16 = N; VGPR0/1 = K + 2*(lane/16)  (fed from wT).
// C/D: VGPR j -> row j (lanes 0-15) / j+8 (lanes 16-31), col = lane%16.
// ---------------------------------------------------------------------------
__global__ void k_lin(const float* __restrict__ vec,
                      const float* __restrict__ wT,
                      const float* __restrict__ b_pad,
                      float* __restrict__ lin) {
    const int lane = threadIdx.x;          // block = 32 (one wave, EXEC all 1s)
    const int nT = blockIdx.x;             // 0..18
    const int mT = blockIdx.y;             // 0..15
    const int hi = lane >> 4;              // 0/1
    const int lo = lane & 15;
    const int m = mT * 16 + lo;
    const int n = nT * 16 + lo;

    const float* __restrict__ ap = vec + (size_t)m * D;   // row of A
    const float* __restrict__ bp = wT  + (size_t)n * D;   // transposed col of B

    v8f c = {};
#pragma unroll 5
    for (int k = 0; k < D; k += 4) {
        const int ka = k + 2 * hi;         // even -> 8B-aligned pair loads
        v2f a;
        a.x = ap[ka];
        a.y = ap[ka + 1];
        v2f bb;
        bb.x = bp[ka];
        bb.y = bp[ka + 1];
        c = __builtin_amdgcn_wmma_f32_16x16x4_f32(
                /*neg_a=*/false, a, /*neg_b=*/false, bb,
                /*c_mod=*/(short)0, c, /*reuse_a=*/false, /*reuse_b=*/false);
    }
    const float bv = b_pad[n];
#pragma unroll
    for (int j = 0; j < 8; ++j) {
        const int mr = mT * 16 + j + 8 * hi;
        lin[mr * NPAD + n] = c[j] + bv;    // padded cols hold bias-free zeros; never read
    }
}

// ---------------------------------------------------------------------------
// Kernel 4 (per hop): fused vec-score -> tanh -> masked softmax -> attention
// readout over L2-resident ext_mem -> add linear.  One block (256 thr) per b.
// ---------------------------------------------------------------------------
__global__ void k_hop(const float* __restrict__ vec_in,
                      const float* __restrict__ w_att,
                      const float* __restrict__ b_att,
                      const int*   __restrict__ ctx_len,
                      const float* __restrict__ mem_score,
                      const float* __restrict__ ext_mem,
                      const float* __restrict__ lin,
                      float* __restrict__ vec_out) {
    __shared__ float red[256];
    __shared__ float alpha[256];
    __shared__ float vs_sh;

    const int b = blockIdx.x;
    const int t = threadIdx.x;

    // vec-dependent score term: dot(vec, w_att[D:2D]) + b_att
    float p = 0.0f;
    if (t < D)        p  = vec_in[b * D + t]       * w_att[D + t];
    if (t + 256 < D)  p += vec_in[b * D + t + 256] * w_att[D + t + 256];
    red[t] = p; __syncthreads();
    for (int s = 128; s; s >>= 1) { if (t < s) red[t] += red[t + s]; __syncthreads(); }
    if (t == 0) vs_sh = red[0] + b_att[0];
    __syncthreads();
    const float vs = vs_sh;

    // tanh score + masked softmax over l = t
    const int  len   = ctx_len[b];
    const bool valid = (t < len);
    const float sc = valid ? tanhf(mem_score[b * L + t] + vs) : -1e9f;
    red[t] = sc; __syncthreads();
    for (int s = 128; s; s >>= 1) { if (t < s) red[t] = fmaxf(red[t], red[t + s]); __syncthreads(); }
    const float mx = red[0]; __syncthreads();
    const float pe = valid ? expf(sc - mx) : 0.0f;
    red[t] = pe; __syncthreads();
    for (int s = 128; s; s >>= 1) { if (t < s) red[t] += red[t + s]; __syncthreads(); }
    const float inv = 1.0f / red[0];
    alpha[t] = pe * inv;
    __syncthreads();

    // attn_out = alpha @ ext_mem[b]   (coalesced streaming from L2) + linear
    float acc0 = 0.0f, acc1 = 0.0f;
    const bool has1 = (t + 256) < D;
    const float* __restrict__ em = ext_mem + (size_t)b * L * D;
    for (int l = 0; l < L; ++l) {
        const float a = alpha[l];
        acc0 += a * em[(size_t)l * D + t];
        if (has1) acc1 += a * em[(size_t)l * D + t + 256];
    }
    vec_out[b * D + t] = acc0 + lin[b * NPAD + t];
    if (has1) vec_out[b * D + t + 256] = acc1 + lin[b * NPAD + t + 256];
}

// ---------------------------------------------------------------------------
// Kernel 5: logits = vec @ w_out + b_out   ([256,300]x[300,3], tiny)
// ---------------------------------------------------------------------------
__global__ void k_out(const float* __restrict__ vec,
                      const float* __restrict__ w_out,
                      const float* __restrict__ b_out,
                      float* __restrict__ out) {
    const int b = blockIdx.x;
    const int lane = threadIdx.x;   // block = 32
    for (int c = 0; c < 3; ++c) {
        float p = 0.0f;
        for (int d = lane; d < D; d += 32)
            p += vec[b * D + d] * w_out[d * 3 + c];
        for (int off = 16; off; off >>= 1) p += __shfl_xor(p, off, 32);
        if (lane == 0) out[b * 3 + c] = p + b_out[c];
    }
}

// ---------------------------------------------------------------------------
extern "C" void kernel_launch(void* const* d_in, const int* in_sizes, int n_in,
                              void* d_out, int out_size, void* d_ws, size_t ws_size,
                              hipStream_t stream) {
    (void)in_sizes; (void)n_in; (void)out_size; (void)ws_size;
    const float* emb    = (const float*)d_in[0];
    const float* w_lin  = (const float*)d_in[1];
    const float* b_lin  = (const float*)d_in[2];
    const float* w_att  = (const float*)d_in[3];
    const float* b_att  = (const float*)d_in[4];
    const float* w_out  = (const float*)d_in[5];
    const float* b_out  = (const float*)d_in[6];
    const int* ctx_ids  = (const int*)d_in[7];
    const int* ctx_len  = (const int*)d_in[8];
    const int* tgt_ids  = (const int*)d_in[9];
    const int* tgt_len  = (const int*)d_in[10];
    const int* tgt_loc  = (const int*)d_in[11];
    float* out = (float*)d_out;

    float* ws        = (float*)d_ws;
    float* ext_mem   = ws;                             // B*L*D  (78.6 MB, L2-resident)
    float* mem_score = ext_mem + (size_t)B * L * D;    // B*L
    float* vecA      = mem_score + (size_t)B * L;      // B*D
    float* vecB      = vecA + (size_t)B * D;           // B*D
    float* lin       = vecB + (size_t)B * D;           // B*NPAD
    float* wT        = lin + (size_t)B * NPAD;         // NPAD*D (transposed, padded)
    float* b_pad     = wT + (size_t)NPAD * D;          // NPAD

    k_prep<<<NPAD, 256, 0, stream>>>(w_lin, b_lin, wT, b_pad);
    k_gather<<<dim3(L / 8, B), 256, 0, stream>>>(emb, w_att, ctx_ids, ctx_len,
                                                 tgt_loc, ext_mem, mem_score);
    k_aspect<<<B, 320, 0, stream>>>(emb, tgt_ids, tgt_len, vecA);

    float* cur = vecA;
    float* nxt = vecB;
    for (int h = 0; h < NHOPS; ++h) {
        k_lin<<<dim3(19, 16), 32, 0, stream>>>(cur, wT, b_pad, lin);
        k_hop<<<B, 256, 0, stream>>>(cur, w_att, b_att, ctx_len, mem_score,
                                     ext_mem, lin, nxt);
        float* tmp = cur; cur = nxt; nxt = tmp;
    }
    k_out<<<B, 32, 0, stream>>>(cur, w_out, b_out, out);
}